// HPUUnifiedAttentionImpl_71141838291324
// MI455X (gfx1250) — compile-verified
//
#include <hip/hip_runtime.h>

#define NUM_HEADS 32
#define KVH       8
#define G         4
#define D         128
#define BQ        32
#define BLOCK_SZ  128
#define MAX_BLKS  32
#define TILE      16
#define WAVES     8
#define THREADS   (WAVES * 32)
#define DP        (D + 4)      // bank-conflict padding: stride % 64 banks == 4
#define SCALE     0.08838834764831845f

typedef float v2f __attribute__((ext_vector_type(2)));
typedef float v8f __attribute__((ext_vector_type(8)));

// ---- LDS layout (float offsets) ----
#define Q_OFF   0
#define O_OFF   (Q_OFF + 16 * DP)          // 4*128 shared output accumulator
#define RM_OFF  (O_OFF + G * D)            // per-wave row max   [WAVES][16]
#define RL_OFF  (RM_OFF + WAVES * 16)      // per-wave row sum   [WAVES][16]
#define K_OFF   (RL_OFF + WAVES * 16)      // per-wave K tile    [WAVES][16][DP]
#define V_OFF   (K_OFF + WAVES * TILE * DP)
#define LDS_FLOATS (V_OFF + WAVES * TILE * DP)

__global__ void __launch_bounds__(THREADS)
paged_attn_decode_wmma(const float* __restrict__ q,
                       const float* __restrict__ knew,
                       const float* __restrict__ vnew,
                       const float* __restrict__ kcache,
                       const float* __restrict__ vcache,
                       const long long* __restrict__ slot_map,
                       const int* __restrict__ btab,
                       const int* __restrict__ ctxlen,
                       float* __restrict__ out)
{
    extern __shared__ float lds[];
    const int b    = blockIdx.x >> 3;   // sequence
    const int kv   = blockIdx.x & 7;    // kv head
    const int tid  = threadIdx.x;
    const int w    = tid >> 5;          // wave id (wave32)
    const int lane = tid & 31;
    const int ln   = lane & 15;
    const int h    = lane >> 4;

    // ---- stage Q (zero-padded to 16 heads) + zero shared output accumulator
    for (int i = tid; i < 16 * D; i += THREADS) {
        const int hh = i >> 7, d = i & 127;
        lds[Q_OFF + hh * DP + d] =
            (hh < G) ? q[((size_t)b * NUM_HEADS + kv * G + hh) * D + d] : 0.0f;
    }
    for (int i = tid; i < G * D; i += THREADS) lds[O_OFF + i] = 0.0f;
    __syncthreads();

    const int ctx = ctxlen[b];
    float* Kl = &lds[K_OFF + w * TILE * DP];
    float* Vl = &lds[V_OFF + w * TILE * DP];

    const v8f vzero = {0.f, 0.f, 0.f, 0.f, 0.f, 0.f, 0.f, 0.f};
    v8f acc[8];
#pragma unroll
    for (int i = 0; i < 8; ++i) acc[i] = vzero;
    float m = -__builtin_inff();
    float l = 0.0f;

    // ---- per-wave token tiles (16 tokens each), online softmax ----
    for (int tile = w; tile * TILE < ctx; tile += WAVES) {
        const int s0   = tile * TILE;
        const int blk  = btab[b * MAX_BLKS + (s0 >> 7)];
        const int tok0 = s0 & (BLOCK_SZ - 1);

        // prefetch next tile's K/V region (speculative, global_prefetch_b8)
        {
            const int nt = tile + WAVES;
            if (nt * TILE < ctx) {
                const int nblk = btab[b * MAX_BLKS + ((nt * TILE) >> 7)];
                const size_t no =
                    (((size_t)nblk * BLOCK_SZ + ((nt * TILE) & (BLOCK_SZ - 1))) * KVH + kv) * D;
                const int r0 = lane >> 2, c0 = (lane & 3) * 32;
                __builtin_prefetch(kcache + no + (size_t)r0 * (KVH * D) + c0, 0, 1);
                __builtin_prefetch(kcache + no + (size_t)(r0 + 8) * (KVH * D) + c0, 0, 1);
                __builtin_prefetch(vcache + no + (size_t)r0 * (KVH * D) + c0, 0, 1);
                __builtin_prefetch(vcache + no + (size_t)(r0 + 8) * (KVH * D) + c0, 0, 1);
            }
        }

        // ---- stage K/V tile into LDS, applying slot_mapping overrides ----
        for (int t = 0; t < TILE; ++t) {
            const long long slot = (long long)blk * BLOCK_SZ + (tok0 + t);
            int rep = -1;
#pragma unroll
            for (int j = 0; j < BQ; ++j)
                if (slot_map[j] == slot) rep = j;   // last match wins (scatter semantics)
            const float* ks;
            const float* vs;
            if (rep >= 0) {
                const size_t o = ((size_t)rep * KVH + kv) * D;
                ks = knew + o; vs = vnew + o;
            } else {
                const size_t o = ((size_t)slot * KVH + kv) * D;
                ks = kcache + o; vs = vcache + o;
            }
            const float4 kr = *(const float4*)(ks + lane * 4);
            const float4 vr = *(const float4*)(vs + lane * 4);
            *(float4*)(Kl + t * DP + lane * 4) = kr;
            *(float4*)(Vl + t * DP + lane * 4) = vr;
        }
        __asm__ volatile("s_wait_dscnt 0" ::: "memory");

        // ---- scores: St = K_tile(16tok x 128d) . Q^T  via 32 chained f32 WMMAs
        v8f st = vzero;
#pragma unroll
        for (int c = 0; c < 32; ++c) {
            const v2f a  = *(const v2f*)(Kl + ln * DP + 4 * c + 2 * h);            // A 16x4
            const v2f bq = *(const v2f*)(&lds[Q_OFF + ln * DP + 4 * c + 2 * h]);   // B 4x16
            st = __builtin_amdgcn_wmma_f32_16x16x4_f32(false, a, false, bq,
                                                       (short)0, st, false, false);
        }

        // ---- scale + mask + online softmax (rows = tokens, cols = heads)
        float sv[8], p[8];
        float tmax = -__builtin_inff();
#pragma unroll
        for (int r = 0; r < 8; ++r) {
            const int tok = s0 + r + 8 * h;
            const float x = (tok < ctx) ? st[r] * SCALE : -__builtin_inff();
            sv[r] = x;
            tmax = fmaxf(tmax, x);
        }
        tmax = fmaxf(tmax, __shfl_xor(tmax, 16, 32));
        const float mn    = fmaxf(m, tmax);
        const float alpha = (mn == -__builtin_inff()) ? 1.0f : __expf(m - mn);
        float ts = 0.0f;
#pragma unroll
        for (int r = 0; r < 8; ++r) {
            const float pe = (sv[r] == -__builtin_inff()) ? 0.0f : __expf(sv[r] - mn);
            p[r] = pe;
            ts += pe;
        }
        ts += __shfl_xor(ts, 16, 32);
        l = l * alpha + ts;
        m = mn;

        // rescale output accumulator by per-head alpha (head = r + 8h per C row)
        float ar[8];
#pragma unroll
        for (int r = 0; r < 8; ++r) ar[r] = __shfl(alpha, (lane < 16) ? r : (r + 8), 32);
#pragma unroll
        for (int dt = 0; dt < 8; ++dt)
#pragma unroll
            for (int r = 0; r < 8; ++r) acc[dt][r] *= ar[r];

        // ---- PV: out(16head x 128d) += P(16head x 16tok) . V(16tok x 128d)
#pragma unroll
        for (int tc = 0; tc < 4; ++tc) {
            // A-layout P chunk: lane needs P[head=ln][tok = 4tc + 2h + j]
            v2f pa;
#pragma unroll
            for (int j = 0; j < 2; ++j) {
                const int tlo = 4 * tc + j, thi = tlo + 2;
                const float vlo = __shfl(p[tlo & 7], ln + ((tlo & 8) ? 16 : 0), 32);
                const float vhi = __shfl(p[thi & 7], ln + ((thi & 8) ? 16 : 0), 32);
                pa[j] = (lane < 16) ? vlo : vhi;
            }
            const float* v0 = Vl + (4 * tc + 2 * h + 0) * DP + ln;
            const float* v1 = Vl + (4 * tc + 2 * h + 1) * DP + ln;
#pragma unroll
            for (int dt = 0; dt < 8; ++dt) {
                v2f bv;
                bv[0] = v0[dt * 16];
                bv[1] = v1[dt * 16];
                acc[dt] = __builtin_amdgcn_wmma_f32_16x16x4_f32(false, pa, false, bv,
                                                               (short)0, acc[dt], false, false);
            }
        }
    }

    // ---- cross-wave flash combine ----
    if (lane < 16) {
        lds[RM_OFF + w * 16 + lane] = m;
        lds[RL_OFF + w * 16 + lane] = l;
    }
    __syncthreads();

    float M = -__builtin_inff();
#pragma unroll
    for (int ww = 0; ww < WAVES; ++ww) M = fmaxf(M, lds[RM_OFF + ww * 16 + ln]);
    const float coef = (M == -__builtin_inff()) ? 0.0f : __expf(m - M);
    float cr[G];
#pragma unroll
    for (int r = 0; r < G; ++r) cr[r] = __shfl(coef, r, 32);

    if (lane < 16) {
#pragma unroll
        for (int dt = 0; dt < 8; ++dt)
#pragma unroll
            for (int r = 0; r < G; ++r)
                atomicAdd(&lds[O_OFF + r * D + dt * 16 + lane], acc[dt][r] * cr[r]);
    }
    __syncthreads();

    // ---- normalize and store ----
    for (int i = tid; i < G * D; i += THREADS) {
        const int g = i >> 7, d = i & 127;
        float Mh = -__builtin_inff();
#pragma unroll
        for (int ww = 0; ww < WAVES; ++ww) Mh = fmaxf(Mh, lds[RM_OFF + ww * 16 + g]);
        float Ls = 0.0f;
#pragma unroll
        for (int ww = 0; ww < WAVES; ++ww) {
            const float mw = lds[RM_OFF + ww * 16 + g];
            if (mw != -__builtin_inff())
                Ls += __expf(mw - Mh) * lds[RL_OFF + ww * 16 + g];
        }
        const float val = (Ls > 0.0f) ? lds[O_OFF + i] / Ls : 0.0f;
        out[(size_t)b * (NUM_HEADS * D) + (size_t)(kv * G + g) * D + d] = val;
    }
}

extern "C" void kernel_launch(void* const* d_in, const int* in_sizes, int n_in,
                              void* d_out, int out_size, void* d_ws, size_t ws_size,
                              hipStream_t stream) {
    const float*     qv = (const float*)d_in[0];
    const float*     kn = (const float*)d_in[1];
    const float*     vn = (const float*)d_in[2];
    const float*     kc = (const float*)d_in[3];
    const float*     vc = (const float*)d_in[4];
    const long long* sm = (const long long*)d_in[5];
    const int*       bt = (const int*)d_in[6];
    const int*       cl = (const int*)d_in[7];
    float*          oud = (float*)d_out;

    const size_t shmem = (size_t)LDS_FLOATS * sizeof(float);  // ~143 KB (320 KB/WGP on CDNA5)
    (void)hipFuncSetAttribute(reinterpret_cast<const void*>(paged_attn_decode_wmma),
                              hipFuncAttributeMaxDynamicSharedMemorySize, (int)shmem);
    paged_attn_decode_wmma<<<BQ * KVH, THREADS, shmem, stream>>>(qv, kn, vn, kc, vc, sm, bt, cl, oud);
}